// GNNRecommender_57372173140420
// MI455X (gfx1250) — compile-verified
//
#include <hip/hip_runtime.h>
#include <math.h>

typedef __attribute__((ext_vector_type(2))) float v2f;
typedef __attribute__((ext_vector_type(8))) float v8f;

#define NUSERS 100000
#define NITEMS 50000
#define NNODES 150000          // NUSERS + NITEMS, multiple of 16 (150000 = 16*9375)
#define BQ     100000          // query pairs, multiple of 16 (100000 = 16*6250)
#define DIM    64
#define H1D    128
#define H2D    128
#define H3D    64

// ---------------- utility kernels ----------------

__global__ void k_zero_f32(float* __restrict__ p, long n) {
    long i = (long)blockIdx.x * blockDim.x + threadIdx.x;
    if (i < n) p[i] = 0.0f;
}

__global__ void k_concat_embed(const float* __restrict__ ut,
                               const float* __restrict__ it,
                               float* __restrict__ x) {
    long i = (long)blockIdx.x * blockDim.x + threadIdx.x;
    const long nu = (long)NUSERS * DIM;
    const long tot = (long)NNODES * DIM;
    if (i < tot) x[i] = (i < nu) ? ut[i] : it[i - nu];
}

__global__ void k_degree(const int* __restrict__ dst, float* __restrict__ deg, int E) {
    int e = blockIdx.x * blockDim.x + threadIdx.x;
    if (e < E) atomicAdd(&deg[dst[e]], 1.0f);
}

__global__ void k_rsqrt(float* __restrict__ deg, int n) {
    int i = blockIdx.x * blockDim.x + threadIdx.x;
    if (i < n) deg[i] = rsqrtf(fmaxf(deg[i], 1.0f));
}

// ---------------- edge scatter: M[dst] += X[src] * dr[src] ----------------
// one thread handles a float4 chunk of one edge; logc = log2(d/4)
__global__ void k_scatter(const float* __restrict__ X, const float* __restrict__ dr,
                          const int* __restrict__ src, const int* __restrict__ dst,
                          float* __restrict__ M, long E, int d, int logc) {
    long tid = (long)blockIdx.x * blockDim.x + threadIdx.x;
    long e = tid >> logc;
    if (e >= E) return;
    int f = ((int)tid & ((1 << logc) - 1)) << 2;
    int s = src[e];
    int t = dst[e];
    float sc = dr[s];
    float4 x = *(const float4*)(X + (size_t)s * d + f);
    float* mp = M + (size_t)t * d + f;
    atomicAdd(mp + 0, x.x * sc);
    atomicAdd(mp + 1, x.y * sc);
    atomicAdd(mp + 2, x.z * sc);
    atomicAdd(mp + 3, x.w * sc);
}

// ---------------- fp32 WMMA GEMM: Y = act((A .* dr[row]) @ W + bias) ----------------
// One wave -> one 16x16 output tile, K-loop in steps of 4 via V_WMMA_F32_16X16X4_F32.
// A-frag (16x4 f32): lane l(0-15) row=l, v0=K0 v1=K1; lane l+16 row=l, v0=K2 v1=K3.
// B-frag (4x16 f32): lane l(0-15) col=l, v0=K0 v1=K1; lane l+16 col=l, v0=K2 v1=K3.
// C/D (16x16 f32): VGPR j = row j (lanes 0-15) / row j+8 (lanes 16-31), col = lane&15.
__global__ void k_gemm_wmma(const float* __restrict__ A, const float* __restrict__ dr,
                            const float* __restrict__ W, const float* __restrict__ bias,
                            float* __restrict__ Y, int nrows, int K, int nout, int relu) {
    int lane = threadIdx.x & 31;
    int wid = blockIdx.x * (blockDim.x >> 5) + (threadIdx.x >> 5);
    int ctiles = nout >> 4;
    int rowTile = wid / ctiles;
    if (rowTile >= (nrows >> 4)) return;           // wave-uniform exit
    int cb = (wid % ctiles) << 4;
    int l = lane & 15;
    int kh = (lane >> 4) << 1;                      // 0 or 2
    int row = (rowTile << 4) + l;
    float drr = dr[row];

    const float* ap = A + (size_t)row * K + kh;
    const float* wp = W + (size_t)kh * nout + cb + l;
    v8f c = {0.f,0.f,0.f,0.f,0.f,0.f,0.f,0.f};
    for (int k = 0; k < K; k += 4) {
        v2f a, b;
        a[0] = ap[k]     * drr;
        a[1] = ap[k + 1] * drr;
        b[0] = wp[(size_t)k * nout];
        b[1] = wp[(size_t)(k + 1) * nout];
        c = __builtin_amdgcn_wmma_f32_16x16x4_f32(false, a, false, b,
                                                  (short)0, c, false, false);
    }
    float bb = bias[cb + l];
    int rbase = (rowTile << 4) + ((lane >> 4) << 3);
#pragma unroll
    for (int j = 0; j < 8; ++j) {
        float v = c[j] + bb;
        if (relu) v = fmaxf(v, 0.0f);
        Y[(size_t)(rbase + j) * nout + cb + l] = v;
    }
}

// ---------------- MLP layer 1: H = relu(concat(X3[u], X3[NU+i]) @ P1 + pb1) ----------------
__global__ void k_mlp1_wmma(const float* __restrict__ X3,
                            const int* __restrict__ uidx, const int* __restrict__ iidx,
                            const float* __restrict__ P1, const float* __restrict__ pb1,
                            float* __restrict__ H) {
    const int K = 2 * H3D;       // 128
    const int nout = 64;
    int lane = threadIdx.x & 31;
    int wid = blockIdx.x * (blockDim.x >> 5) + (threadIdx.x >> 5);
    int ctiles = nout >> 4;      // 4
    int rowTile = wid / ctiles;
    if (rowTile >= (BQ >> 4)) return;              // wave-uniform exit
    int cb = (wid % ctiles) << 4;
    int l = lane & 15;
    int kh = (lane >> 4) << 1;
    int row = (rowTile << 4) + l;

    const float* urow = X3 + (size_t)uidx[row] * H3D;
    const float* irow = X3 + (size_t)(NUSERS + iidx[row]) * H3D;
    const float* wp = P1 + (size_t)kh * nout + cb + l;
    v8f c = {0.f,0.f,0.f,0.f,0.f,0.f,0.f,0.f};
    for (int k = 0; k < K; k += 4) {
        int k0 = k + kh;                            // even, so k0 and k0+1 same half
        const float* zr = (k0 < H3D) ? urow : (irow - H3D);
        v2f a, b;
        a[0] = zr[k0];
        a[1] = zr[k0 + 1];
        b[0] = wp[(size_t)k * nout];
        b[1] = wp[(size_t)(k + 1) * nout];
        c = __builtin_amdgcn_wmma_f32_16x16x4_f32(false, a, false, b,
                                                  (short)0, c, false, false);
    }
    float bb = pb1[cb + l];
    int rbase = (rowTile << 4) + ((lane >> 4) << 3);
#pragma unroll
    for (int j = 0; j < 8; ++j) {
        float v = fmaxf(c[j] + bb, 0.0f);
        H[(size_t)(rbase + j) * nout + cb + l] = v;
    }
}

// ---------------- MLP layers 2+3: out = sigmoid(relu(H@P2+pb2) @ P3 + pb3) ----------------
__global__ void k_mlp23(const float* __restrict__ H,
                        const float* __restrict__ P2, const float* __restrict__ pb2,
                        const float* __restrict__ P3, const float* __restrict__ pb3,
                        float* __restrict__ out) {
    __shared__ float sW[64 * 32];
    __shared__ float sP3[32];
    __shared__ float sB2[32];
    for (int i = threadIdx.x; i < 64 * 32; i += blockDim.x) sW[i] = P2[i];
    if (threadIdx.x < 32) {
        sP3[threadIdx.x] = P3[threadIdx.x];
        sB2[threadIdx.x] = pb2[threadIdx.x];
    }
    __syncthreads();
    int r = blockIdx.x * blockDim.x + threadIdx.x;
    if (r >= BQ) return;
    float h[64];
    const float* hp = H + (size_t)r * 64;
#pragma unroll
    for (int k = 0; k < 64; ++k) h[k] = hp[k];
    float acc = pb3[0];
#pragma unroll 4
    for (int j = 0; j < 32; ++j) {
        float s = sB2[j];
#pragma unroll
        for (int k = 0; k < 64; ++k) s = fmaf(h[k], sW[k * 32 + j], s);
        s = fmaxf(s, 0.0f);
        acc = fmaf(s, sP3[j], acc);
    }
    out[r] = 1.0f / (1.0f + expf(-acc));
}

// ---------------- host orchestration ----------------

static inline int nblk(long n, int t) { return (int)((n + t - 1) / t); }

static void launch_gemm(const float* A, const float* dr, const float* W, const float* b,
                        float* Y, int nrows, int K, int nout, int relu, hipStream_t s) {
    long waves = (long)(nrows / 16) * (nout / 16);
    k_gemm_wmma<<<nblk(waves * 32, 256), 256, 0, s>>>(A, dr, W, b, Y, nrows, K, nout, relu);
}

extern "C" void kernel_launch(void* const* d_in, const int* in_sizes, int n_in,
                              void* d_out, int out_size, void* d_ws, size_t ws_size,
                              hipStream_t stream) {
    const float* user_table = (const float*)d_in[0];
    const float* item_table = (const float*)d_in[1];
    const float* W1 = (const float*)d_in[2];  const float* b1  = (const float*)d_in[3];
    const float* W2 = (const float*)d_in[4];  const float* b2  = (const float*)d_in[5];
    const float* W3 = (const float*)d_in[6];  const float* b3  = (const float*)d_in[7];
    const float* P1 = (const float*)d_in[8];  const float* pb1 = (const float*)d_in[9];
    const float* P2 = (const float*)d_in[10]; const float* pb2 = (const float*)d_in[11];
    const float* P3 = (const float*)d_in[12]; const float* pb3 = (const float*)d_in[13];
    const int* src  = (const int*)d_in[14];
    const int* dst  = (const int*)d_in[15];
    const int* uidx = (const int*)d_in[16];
    const int* iidx = (const int*)d_in[17];
    long E = in_sizes[14];                      // 2*E0 = 2.4M directed edges

    float* ws = (float*)d_ws;
    size_t stride = (size_t)NNODES * 128;
    float* bufA = ws;                           // N x 128
    float* bufB = ws + stride;                  // N x 128 (scatter accumulator)
    float* bufC = ws + 2 * stride;              // N x 128
    float* deg  = ws + 3 * stride;              // N
    float* h1   = deg + NNODES;                 // B x 64
    float* out  = (float*)d_out;

    // degree -> deg_rsqrt
    k_zero_f32<<<nblk(NNODES, 256), 256, 0, stream>>>(deg, NNODES);
    k_degree<<<nblk(E, 256), 256, 0, stream>>>(dst, deg, (int)E);
    k_rsqrt<<<nblk(NNODES, 256), 256, 0, stream>>>(deg, NNODES);

    // X0 = concat(user_table, item_table)  -> bufA [N x 64]
    k_concat_embed<<<nblk((long)NNODES * DIM, 256), 256, 0, stream>>>(user_table, item_table, bufA);

    // layer 1: d_in=64 -> d_out=128, relu
    k_zero_f32<<<nblk((long)NNODES * 64, 256), 256, 0, stream>>>(bufB, (long)NNODES * 64);
    k_scatter<<<nblk(E * 16, 256), 256, 0, stream>>>(bufA, deg, src, dst, bufB, E, 64, 4);
    launch_gemm(bufB, deg, W1, b1, bufC, NNODES, 64, 128, 1, stream);

    // layer 2: 128 -> 128, relu
    k_zero_f32<<<nblk((long)NNODES * 128, 256), 256, 0, stream>>>(bufB, (long)NNODES * 128);
    k_scatter<<<nblk(E * 32, 256), 256, 0, stream>>>(bufC, deg, src, dst, bufB, E, 128, 5);
    launch_gemm(bufB, deg, W2, b2, bufA, NNODES, 128, 128, 1, stream);

    // layer 3: 128 -> 64, no relu
    k_zero_f32<<<nblk((long)NNODES * 128, 256), 256, 0, stream>>>(bufB, (long)NNODES * 128);
    k_scatter<<<nblk(E * 32, 256), 256, 0, stream>>>(bufA, deg, src, dst, bufB, E, 128, 5);
    launch_gemm(bufB, deg, W3, b3, bufC, NNODES, 128, 64, 0, stream);

    // MLP layer 1 (gathered concat rows), WMMA: [B x 128] @ [128 x 64] -> h1
    long waves1 = (long)(BQ / 16) * (64 / 16);
    k_mlp1_wmma<<<nblk(waves1 * 32, 256), 256, 0, stream>>>(bufC, uidx, iidx, P1, pb1, h1);

    // MLP layers 2+3 + sigmoid
    k_mlp23<<<nblk(BQ, 256), 256, 0, stream>>>(h1, P2, pb2, P3, pb3, out);
}